// KromHC_13606456394139
// MI455X (gfx1250) — compile-verified
//
#include <hip/hip_runtime.h>
#include <math.h>

// ---------------------------------------------------------------------------
// Problem constants (from reference): S=8, K=3, T=6, D=2048, B=2, N=1024
//   residuals: (B*S, N, D) f32 ; out: (B*S, N, D) f32
//   tokens = B*N = 2048 ; per-token vector length S*D = 16384
// ---------------------------------------------------------------------------
#define NTOK   2048
#define SD     16384
#define DD     2048
#define NN     1024
#define TPB    8          // tokens per workgroup in pass 1

typedef __attribute__((ext_vector_type(2))) float v2f;
typedef __attribute__((ext_vector_type(8))) float v8f;

// ---------------------------------------------------------------------------
// Pass 1: per-token reductions -> 8x8 mixing matrix M, stored to workspace.
//   red[c]   = sum_i r[i]*(gamma[i]+1)*Wc[i][c]   (c = 0..13 W_alpha, 14..21 W_beta)
//   red[22]  = sum_i r[i]^2
//   coeffs   = red * rsqrt(max(sumsq,1e-24)) * sqrt(S*D)
//   M[t][s]  = hres[s][t] + beta[t]*w_pre[s]
// One WG = 256 threads handles TPB=8 consecutive tokens; each thread reads a
// W row once and reuses it for all 8 tokens (register accumulators).
// ---------------------------------------------------------------------------
__global__ __launch_bounds__(256)
void kromhc_coeff_kernel(const float* __restrict__ res,
                         const float* __restrict__ gamma,
                         const float* __restrict__ st_alpha,
                         const float* __restrict__ Wa,
                         const float* __restrict__ pre_s_p,
                         const float* __restrict__ res_s_p,
                         const float* __restrict__ st_beta,
                         const float* __restrict__ Wb,
                         const float* __restrict__ hps_p,
                         float* __restrict__ Mws)
{
    __shared__ float lds_red[23 * 256];
    __shared__ float lds_fin[TPB][23];

    const int tid  = threadIdx.x;
    const int tok0 = blockIdx.x * TPB;          // 8 tokens, all same b
    const int b    = tok0 >> 10;

    float acc[TPB][23];
    #pragma unroll
    for (int t = 0; t < TPB; ++t)
        #pragma unroll
        for (int c = 0; c < 23; ++c) acc[t][c] = 0.0f;

    for (int it = 0; it < SD / 256; ++it) {
        const int i  = it * 256 + tid;          // element index in [0, 16384)
        const int se = i >> 11;                 // s
        const int d  = i & (DD - 1);
        const float g1 = gamma[i] + 1.0f;
        float wrow[22];
        #pragma unroll
        for (int c = 0; c < 14; ++c) wrow[c]      = Wa[(size_t)i * 14 + c];
        #pragma unroll
        for (int c = 0; c < 8;  ++c) wrow[14 + c] = Wb[(size_t)i * 8 + c];
        const size_t base = ((size_t)(b * 8 + se) * NN) * DD + d;
        #pragma unroll
        for (int t = 0; t < TPB; ++t) {
            const int n = (tok0 + t) & (NN - 1);
            const float r = res[base + (size_t)n * DD];
            acc[t][22] = fmaf(r, r, acc[t][22]);
            const float w = r * g1;
            #pragma unroll
            for (int c = 0; c < 22; ++c) acc[t][c] = fmaf(w, wrow[c], acc[t][c]);
        }
    }

    // Deterministic reduction (fixed order): 256 partials -> 23 sums per token.
    for (int t = 0; t < TPB; ++t) {
        #pragma unroll
        for (int c = 0; c < 23; ++c) lds_red[c * 256 + tid] = acc[t][c];
        __syncthreads();
        if (tid < 23) {
            float s = 0.0f;
            for (int j = 0; j < 256; ++j) s += lds_red[tid * 256 + j];
            lds_fin[t][tid] = s;
        }
        __syncthreads();
    }

    if (tid < TPB) {
        const int tok = tok0 + tid;
        float red[23];
        #pragma unroll
        for (int c = 0; c < 23; ++c) red[c] = lds_fin[tid][c];

        const float inv   = rsqrtf(fmaxf(red[22], 1e-24f));
        const float scale = inv * 128.0f;                 // sqrt(S*D) = 128
        const float pre_s = pre_s_p[0];
        const float rs    = res_s_p[0];
        const float hps   = hps_p[0];

        // w_pre = softmax(pre_s * dyn_pre + static_alpha[:8])
        float logit[8], mx = -1e30f;
        #pragma unroll
        for (int s = 0; s < 8; ++s) {
            logit[s] = pre_s * (scale * red[s]) + st_alpha[s];
            mx = fmaxf(mx, logit[s]);
        }
        float wpre[8], den = 0.0f;
        #pragma unroll
        for (int s = 0; s < 8; ++s) { wpre[s] = __expf(logit[s] - mx); den += wpre[s]; }
        const float rden = 1.0f / den;
        #pragma unroll
        for (int s = 0; s < 8; ++s) wpre[s] *= rden;

        // p[k] = softmax over pairs, take component 0
        float p[3];
        #pragma unroll
        for (int k = 0; k < 3; ++k) {
            const float c0 = rs * (scale * red[8 + 2 * k])     + st_alpha[8 + 2 * k];
            const float c1 = rs * (scale * red[8 + 2 * k + 1]) + st_alpha[8 + 2 * k + 1];
            p[k] = 1.0f / (1.0f + __expf(c1 - c0));
        }

        // beta = sigmoid(hps * (normed @ W_beta) + static_beta)
        float beta[8];
        #pragma unroll
        for (int t = 0; t < 8; ++t)
            beta[t] = 1.0f / (1.0f + __expf(-(hps * (scale * red[14 + t]) + st_beta[t])));

        // hres[s][t] = prod_k (s_k==t_k ? p[k] : 1-p[k]);  M[t][s] = hres[s][t] + beta[t]*wpre[s]
        #pragma unroll
        for (int t = 0; t < 8; ++t)
            #pragma unroll
            for (int s = 0; s < 8; ++s) {
                float h = 1.0f;
                #pragma unroll
                for (int k = 0; k < 3; ++k) {
                    const int bs = (s >> (2 - k)) & 1;
                    const int bt = (t >> (2 - k)) & 1;
                    h *= (bs == bt) ? p[k] : (1.0f - p[k]);
                }
                Mws[(size_t)tok * 64 + t * 8 + s] = h + beta[t] * wpre[s];
            }
    }
}

// ---------------------------------------------------------------------------
// Pass 2: out = M @ r per token, via V_WMMA_F32_16X16X4_F32.
// Two tokens per workgroup -> block-diagonal 16x16 A (K=16 as 4 chained
// k=4 WMMAs). C/D layout: VGPR j, lanes 0-15 = row j (token0), lanes 16-31 =
// row j+8 (token1) -> all 32 lanes store useful data. Output stores are
// non-temporal so the 128 MiB write stream doesn't evict L2-resident r.
// ---------------------------------------------------------------------------
__global__ __launch_bounds__(256)
void kromhc_mix_kernel(const float* __restrict__ res,
                       const float* __restrict__ Mws,
                       float* __restrict__ out)
{
    const int pair = blockIdx.x;         // tokens 2*pair, 2*pair+1 (same b)
    const int tid  = threadIdx.x;
    const int lane = tid & 31;
    const int wave = tid >> 5;           // 8 waves, each 256 columns
    const int half = lane >> 4;
    const int l15  = lane & 15;

    // ---- A matrix (block-diagonal Mbig), ISA layout: lane = M row,
    //      VGPR v + lane-half h -> K = 4i + 2h + v for WMMA slice i.
    const int row  = l15;                // M row 0..15
    const int rtok = row >> 3;
    const int t    = row & 7;
    v2f a[4];
    #pragma unroll
    for (int i = 0; i < 4; ++i) {
        float av[2];
        #pragma unroll
        for (int v = 0; v < 2; ++v) {
            const int kk = 4 * i + 2 * half + v;
            av[v] = 0.0f;
            if ((kk >> 3) == rtok) {
                const int tok = pair * 2 + rtok;
                av[v] = Mws[(size_t)tok * 64 + t * 8 + (kk & 7)];
            }
        }
        a[i].x = av[0]; a[i].y = av[1];
    }

    // ---- per-lane B row offsets: j = 2i+v -> K = 4i + 2*half + v
    size_t rowOff[8];
    #pragma unroll
    for (int i = 0; i < 4; ++i)
        #pragma unroll
        for (int v = 0; v < 2; ++v) {
            const int K   = 4 * i + 2 * half + v;
            const int tok = pair * 2 + (K >> 3);
            const int bb  = tok >> 10, nn = tok & (NN - 1);
            const int s   = K & 7;
            rowOff[2 * i + v] = ((size_t)((bb * 8 + s) * NN + nn)) * DD;
        }

    // ---- per-lane output row offsets (lane half selects token)
    const int otok = pair * 2 + half;
    const int ob = otok >> 10, on = otok & (NN - 1);
    size_t outOff[8];
    #pragma unroll
    for (int j = 0; j < 8; ++j)
        outOff[j] = ((size_t)((ob * 8 + j) * NN + on)) * DD;

    // ---- 16 chunks of 16 columns per wave; 4 WMMAs (K=16) per chunk
    for (int ci = 0; ci < 16; ++ci) {
        const int col = (wave * 16 + ci) * 16 + l15;
        v8f c = {};
        #pragma unroll
        for (int i = 0; i < 4; ++i) {
            v2f bv;
            bv.x = res[rowOff[2 * i + 0] + col];
            bv.y = res[rowOff[2 * i + 1] + col];
            c = __builtin_amdgcn_wmma_f32_16x16x4_f32(
                    /*neg_a=*/false, a[i], /*neg_b=*/false, bv,
                    /*c_mod=*/(short)0, c, /*reuse_a=*/false, /*reuse_b=*/false);
        }
        #pragma unroll
        for (int j = 0; j < 8; ++j)
            __builtin_nontemporal_store(c[j], &out[outOff[j] + col]);
    }
}

// ---------------------------------------------------------------------------
extern "C" void kernel_launch(void* const* d_in, const int* in_sizes, int n_in,
                              void* d_out, int out_size, void* d_ws, size_t ws_size,
                              hipStream_t stream)
{
    const float* residuals  = (const float*)d_in[0];
    const float* gamma      = (const float*)d_in[1];
    const float* st_alpha   = (const float*)d_in[2];
    const float* W_alpha    = (const float*)d_in[3];
    const float* pre_scale  = (const float*)d_in[4];
    const float* res_scale  = (const float*)d_in[5];
    const float* st_beta    = (const float*)d_in[6];
    const float* W_beta     = (const float*)d_in[7];
    const float* hps        = (const float*)d_in[8];
    float*       out        = (float*)d_out;
    float*       Mws        = (float*)d_ws;          // 2048 * 64 floats = 512 KB

    kromhc_coeff_kernel<<<NTOK / TPB, 256, 0, stream>>>(
        residuals, gamma, st_alpha, W_alpha, pre_scale, res_scale,
        st_beta, W_beta, hps, Mws);

    kromhc_mix_kernel<<<NTOK / 2, 256, 0, stream>>>(residuals, Mws, out);
}